// TransitionDownBlock_66254165508549
// MI455X (gfx1250) — compile-verified
//
#include <hip/hip_runtime.h>
#include <hip/hip_bf16.h>

#define GRID_SZ 0.05f
#define BN_EPS  1e-5f

constexpr int  Bn     = 8;
constexpr int  Np     = 65536;
constexpr int  CIN    = 64;
constexpr int  COUT   = 128;
constexpr int  VOXCAP = 8192;                 // > 20^3 voxel ids
constexpr long MROWS  = (long)Bn * Np;        // 524288

typedef float v2f __attribute__((ext_vector_type(2)));
typedef float v8f __attribute__((ext_vector_type(8)));

// ---------------- workspace layout (bytes) ----------------
constexpr size_t OFF_SUM   = 0;                                  // float[128]
constexpr size_t OFF_SUMSQ = OFF_SUM   + COUT * 4;               // float[128]
constexpr size_t OFF_SCALE = OFF_SUMSQ + COUT * 4;               // float[128]
constexpr size_t OFF_SHIFT = OFF_SCALE + COUT * 4;               // float[128]
constexpr size_t OFF_MINB  = OFF_SHIFT + COUT * 4;               // uint [8][4]
constexpr size_t OFF_MAXV  = OFF_MINB + Bn * 4 * 4;              // int  [8][4]
constexpr size_t OFF_NC    = OFF_MAXV + Bn * 4 * 4;              // int  [8]
constexpr size_t OFF_RANK  = 4096;                               // int  [8][VOXCAP]
constexpr size_t OFF_VID   = OFF_RANK + (size_t)Bn * VOXCAP * 4; // int  [B*N]
constexpr size_t OFF_CNT   = OFF_VID + (size_t)MROWS * 4;        // float[8][VOXCAP]
constexpr size_t OFF_PSUM  = OFF_CNT + (size_t)Bn * VOXCAP * 4;  // float[8][VOXCAP][3]
constexpr size_t OFF_FMAX  = OFF_PSUM + (size_t)Bn * VOXCAP * 3 * 4; // uint[8][VOXCAP][128]
constexpr size_t WS_TOTAL  = OFF_FMAX + (size_t)Bn * VOXCAP * COUT * 4;

// ---------------- utility kernels ----------------
__global__ void k_zero(uint4* p, long n) {
  long i = (long)blockIdx.x * blockDim.x + threadIdx.x;
  long stride = (long)gridDim.x * blockDim.x;
  uint4 z = {0u, 0u, 0u, 0u};
  for (; i < n; i += stride) p[i] = z;
}

__global__ void k_setmin(unsigned* minb) {
  int i = threadIdx.x;
  if (i < Bn * 4) minb[i] = 0x7F800000u;   // +inf bits
}

__global__ void k_minred(const float* __restrict__ xyz, unsigned* __restrict__ minb) {
  long p = (long)blockIdx.x * 256 + threadIdx.x;
  if (p >= MROWS) return;
  int b = (int)(p >> 16);
  const float* q = xyz + p * 3;
  #pragma unroll
  for (int d = 0; d < 3; ++d)
    atomicMin(&minb[b * 4 + d], __float_as_uint(q[d]));  // exact for floats >= 0
}

__global__ void k_vmax(const float* __restrict__ xyz, const unsigned* __restrict__ minb,
                       int* __restrict__ maxv) {
  long p = (long)blockIdx.x * 256 + threadIdx.x;
  if (p >= MROWS) return;
  int b = (int)(p >> 16);
  const float* q = xyz + p * 3;
  #pragma unroll
  for (int d = 0; d < 3; ++d) {
    float mn = __uint_as_float(minb[b * 4 + d]);
    int v = (int)floorf((q[d] - mn) / GRID_SZ);
    atomicMax(&maxv[b * 4 + d], v);
  }
}

__global__ void k_vidmark(const float* __restrict__ xyz, const unsigned* __restrict__ minb,
                          const int* __restrict__ maxv, int* __restrict__ vid,
                          int* __restrict__ rank) {
  long p = (long)blockIdx.x * 256 + threadIdx.x;
  if (p >= MROWS) return;
  int b = (int)(p >> 16);
  const float* q = xyz + p * 3;
  int v[3];
  #pragma unroll
  for (int d = 0; d < 3; ++d) {
    float mn = __uint_as_float(minb[b * 4 + d]);
    v[d] = (int)floorf((q[d] - mn) / GRID_SZ);
  }
  int mx1 = maxv[b * 4 + 1] + 1;
  int mx2 = maxv[b * 4 + 2] + 1;
  int id = (v[0] * mx1 + v[1]) * mx2 + v[2];
  id = min(max(id, 0), VOXCAP - 1);
  vid[p] = id;
  rank[(size_t)b * VOXCAP + id] = 1;   // idempotent occupancy mark
}

// per-batch exclusive scan of occupancy -> unique-sorted cluster index
__global__ __launch_bounds__(1024) void k_scan(int* __restrict__ rank, int* __restrict__ ncArr) {
  int b = blockIdx.x;
  int* r = rank + (size_t)b * VOXCAP;
  __shared__ int ts[1024];
  int tid = threadIdx.x;
  int base = tid * 8;
  int v[8];
  int s = 0;
  #pragma unroll
  for (int i = 0; i < 8; ++i) { v[i] = r[base + i]; s += v[i]; }
  ts[tid] = s;
  __syncthreads();
  for (int off = 1; off < 1024; off <<= 1) {
    int x = ts[tid];
    int y = (tid >= off) ? ts[tid - off] : 0;
    __syncthreads();
    ts[tid] = x + y;
    __syncthreads();
  }
  int excl = ts[tid] - s;   // exclusive prefix of this thread's chunk
  #pragma unroll
  for (int i = 0; i < 8; ++i) { int f = v[i]; r[base + i] = excl; excl += f; }
  if (tid == 1023) ncArr[b] = ts[1023];
}

// ---------------- fused WMMA GEMM (fp32 16x16x4, K=64 chained) ----------------
// STATS=true : accumulate per-channel sum/sumsq of h = X@W + b
// STATS=false: y = relu(h*scale+shift') scattered via bitwise atomicMax per cluster
template <bool STATS>
__global__ __launch_bounds__(256) void k_gemm(
    const float* __restrict__ X, const float* __restrict__ Wm,
    const float* __restrict__ bias,
    const float* __restrict__ scaleA, const float* __restrict__ shiftA,
    const int* __restrict__ vidA, const int* __restrict__ rankA,
    float* __restrict__ gSum, float* __restrict__ gSq,
    unsigned* __restrict__ fmaxAcc) {
  __shared__ float Wl[CIN * COUT];   // 32 KB: whole W, broadcast to 8 waves
  __shared__ float bL[COUT];
  __shared__ float scL[COUT];
  __shared__ float shL[COUT];
  __shared__ float sSum[COUT];
  __shared__ float sSq[COUT];

  int tid = threadIdx.x;
  for (int i = tid; i < CIN * COUT; i += 256) Wl[i] = Wm[i];
  if (tid < COUT) {
    bL[tid] = bias[tid];
    if (!STATS) { scL[tid] = scaleA[tid]; shL[tid] = shiftA[tid]; }
    else        { sSum[tid] = 0.f; sSq[tid] = 0.f; }
  }
  __syncthreads();

  int wv   = tid >> 5;
  int lane = tid & 31;
  int m16  = lane & 15;       // A: M row for this lane
  int kh   = lane >> 4;       // A: K-pair select (K+0/1 vs K+2/3)
  long row0  = (long)blockIdx.x * 128 + wv * 16;   // 16 rows per wave
  int  batch = (int)(row0 >> 16);                  // blocks never straddle clouds

  // A tile 16x64 fp32 fully in registers: 16 chained 16x4 operands (v2f each)
  v2f a[16];
  const float* Xw = X + (row0 + m16) * CIN;
  #pragma unroll
  for (int s = 0; s < 16; ++s)
    a[s] = *(const v2f*)(Xw + s * 4 + kh * 2);

  int cl[8];
  if (!STATS) {
    #pragma unroll
    for (int i = 0; i < 8; ++i) {      // VGPR i holds M=i (lanes<16) / M=8+i (lanes>=16)
      long r = row0 + i + kh * 8;
      cl[i] = rankA[(size_t)batch * VOXCAP + vidA[r]];
    }
  }

  #pragma unroll
  for (int t = 0; t < COUT / 16; ++t) {
    int n = (t << 4) + m16;            // output column for this lane
    v8f c = {};
    #pragma unroll
    for (int s = 0; s < 16; ++s) {
      int kr = s * 4 + kh * 2;
      v2f bb;
      bb.x = Wl[kr * COUT + n];
      bb.y = Wl[(kr + 1) * COUT + n];
      c = __builtin_amdgcn_wmma_f32_16x16x4_f32(false, a[s], false, bb,
                                                (short)0, c, false, false);
    }
    if (STATS) {
      float bn = bL[n];
      float s1 = 0.f, s2 = 0.f;
      #pragma unroll
      for (int i = 0; i < 8; ++i) { float h = c[i] + bn; s1 += h; s2 += h * h; }
      atomicAdd(&sSum[n], s1);
      atomicAdd(&sSq[n], s2);
    } else {
      float sc = scL[n], sh = shL[n];  // bias already folded into sh
      #pragma unroll
      for (int i = 0; i < 8; ++i) {
        float y = fmaxf(c[i] * sc + sh, 0.f);   // >= 0 -> uint-bits max is exact
        atomicMax(&fmaxAcc[((size_t)batch * VOXCAP + cl[i]) * COUT + n],
                  __float_as_uint(y));
      }
    }
  }

  if (STATS) {
    __syncthreads();
    if (tid < COUT) {
      atomicAdd(&gSum[tid], sSum[tid]);
      atomicAdd(&gSq[tid], sSq[tid]);
    }
  }
}

__global__ void k_statfin(const float* __restrict__ gSum, const float* __restrict__ gSq,
                          const float* __restrict__ bias, const float* __restrict__ gamma,
                          const float* __restrict__ beta, float* __restrict__ scale,
                          float* __restrict__ shift) {
  int c = threadIdx.x;
  if (c < COUT) {
    float invN = 1.0f / (float)MROWS;
    float mu  = gSum[c] * invN;
    float var = gSq[c] * invN - mu * mu;   // biased, torch-BN style
    float sc  = gamma[c] * rsqrtf(var + BN_EPS);
    scale[c] = sc;
    shift[c] = (bias[c] - mu) * sc + beta[c];   // fold bias + BN shift
  }
}

__global__ void k_xyz(const float* __restrict__ xyz, const int* __restrict__ vid,
                      const int* __restrict__ rank, float* __restrict__ psum,
                      float* __restrict__ cnt) {
  long p = (long)blockIdx.x * 256 + threadIdx.x;
  if (p >= MROWS) return;
  int b = (int)(p >> 16);
  int cl = rank[(size_t)b * VOXCAP + vid[p]];
  const float* q = xyz + p * 3;
  size_t base = (size_t)b * VOXCAP + cl;
  atomicAdd(&psum[base * 3 + 0], q[0]);
  atomicAdd(&psum[base * 3 + 1], q[1]);
  atomicAdd(&psum[base * 3 + 2], q[2]);
  atomicAdd(&cnt[base], 1.0f);
}

__global__ void k_out(const float* __restrict__ psum, const float* __restrict__ cnt,
                      const unsigned* __restrict__ fmaxAcc, float* __restrict__ out, int m) {
  long ptsTot = (long)Bn * m * 3;
  long tot = ptsTot + (long)Bn * m * COUT;
  long stride = (long)gridDim.x * blockDim.x;
  for (long i = (long)blockIdx.x * blockDim.x + threadIdx.x; i < tot; i += stride) {
    if (i < ptsTot) {
      long b = i / ((long)m * 3);
      long rem = i - b * (long)m * 3;
      long c = rem / 3;
      int  d = (int)(rem % 3);
      size_t base = (size_t)b * VOXCAP + c;
      out[i] = psum[base * 3 + d] / cnt[base];
    } else {
      long j = i - ptsTot;
      long b = j / ((long)m * COUT);
      long rem = j - b * (long)m * COUT;
      long c = rem / COUT;
      int ch = (int)(rem % COUT);
      out[i] = __uint_as_float(fmaxAcc[((size_t)b * VOXCAP + c) * COUT + ch]);
    }
  }
}

// ---------------- launcher ----------------
extern "C" void kernel_launch(void* const* d_in, const int* in_sizes, int n_in,
                              void* d_out, int out_size, void* d_ws, size_t ws_size,
                              hipStream_t stream) {
  const float* xyz   = (const float*)d_in[0];
  const float* feat  = (const float*)d_in[1];
  const float* Wm    = (const float*)d_in[2];
  const float* bias  = (const float*)d_in[3];
  const float* gamma = (const float*)d_in[4];
  const float* beta  = (const float*)d_in[5];
  float* out = (float*)d_out;

  char* ws = (char*)d_ws;
  float*    gSum  = (float*)(ws + OFF_SUM);
  float*    gSq   = (float*)(ws + OFF_SUMSQ);
  float*    scale = (float*)(ws + OFF_SCALE);
  float*    shift = (float*)(ws + OFF_SHIFT);
  unsigned* minb  = (unsigned*)(ws + OFF_MINB);
  int*      maxv  = (int*)(ws + OFF_MAXV);
  int*      nc    = (int*)(ws + OFF_NC);
  int*      rank  = (int*)(ws + OFF_RANK);
  int*      vid   = (int*)(ws + OFF_VID);
  float*    cnt   = (float*)(ws + OFF_CNT);
  float*    psum  = (float*)(ws + OFF_PSUM);
  unsigned* fmaxA = (unsigned*)(ws + OFF_FMAX);

  size_t zbytes = WS_TOTAL < ws_size ? WS_TOTAL : ws_size;
  k_zero<<<4096, 256, 0, stream>>>((uint4*)ws, (long)(zbytes / 16));
  k_setmin<<<1, 64, 0, stream>>>(minb);

  long pb = (MROWS + 255) / 256;   // 2048 blocks for per-point kernels
  k_minred<<<pb, 256, 0, stream>>>(xyz, minb);
  k_vmax<<<pb, 256, 0, stream>>>(xyz, minb, maxv);
  k_vidmark<<<pb, 256, 0, stream>>>(xyz, minb, maxv, vid, rank);
  k_scan<<<Bn, 1024, 0, stream>>>(rank, nc);

  long gemmBlocks = MROWS / 128;   // 4096 blocks x 8 waves x 16 rows
  k_gemm<true><<<gemmBlocks, 256, 0, stream>>>(feat, Wm, bias, scale, shift,
                                               vid, rank, gSum, gSq, fmaxA);
  k_statfin<<<1, 128, 0, stream>>>(gSum, gSq, bias, gamma, beta, scale, shift);
  k_gemm<false><<<gemmBlocks, 256, 0, stream>>>(feat, Wm, bias, scale, shift,
                                                vid, rank, gSum, gSq, fmaxA);

  k_xyz<<<pb, 256, 0, stream>>>(xyz, vid, rank, psum, cnt);

  int m = out_size / (Bn * (3 + COUT));   // harness sized d_out from the true m
  k_out<<<4096, 256, 0, stream>>>(psum, cnt, fmaxA, out, m);
}